// WeightedBranchRoute_55241869361852
// MI455X (gfx1250) — compile-verified
//
#include <hip/hip_runtime.h>

typedef float v2f __attribute__((ext_vector_type(2)));
typedef float v4f __attribute__((ext_vector_type(4)));
typedef float v8f __attribute__((ext_vector_type(8)));

constexpr int N_TOKENS = 16384;
constexpr int D_MODEL  = 2048;
constexpr int D4       = D_MODEL / 4;   // float4 chunks per row
constexpr int TILE_M   = 16;            // tokens per wave (WMMA M)

__global__ __launch_bounds__(32)
void WeightedBranchRoute_kernel(const float* __restrict__ x,
                                const float* __restrict__ gw,   // (D_MODEL, 2) row-major
                                const float* __restrict__ gb,   // (2,)
                                float* __restrict__ outPre,
                                float* __restrict__ outPost)
{
    const int lane   = threadIdx.x & 31;
    const int token0 = blockIdx.x * TILE_M;
    const int m      = lane & 15;   // A: row M ; B: col N
    const int kh     = lane >> 4;   // K half select (0 -> K,K+1 ; 1 -> K+2,K+3)

    // ---- Phase 1: gate GEMM tile via V_WMMA_F32_16X16X4_F32 ----
    const float* xrow  = x + (size_t)(token0 + m) * D_MODEL;
    const float  bmask = (m < 2) ? 1.0f : 0.0f;            // only N=0,1 columns exist
    const float* gcol  = gw + ((m < 2) ? m : 0);           // safe clamped column

    v8f acc = {};
#pragma unroll 4
    for (int k = 0; k < D_MODEL; k += 4) {
        const int kk = k + 2 * kh;
        v2f a = *(const v2f*)(xrow + kk);                  // global_load_b64, aligned
        v2f b;
        b.x = bmask * gcol[(size_t)kk * 2];
        b.y = bmask * gcol[(size_t)(kk + 1) * 2];
        acc = __builtin_amdgcn_wmma_f32_16x16x4_f32(
            false, a, false, b, (short)0, acc, false, false);
    }

    // ---- Phase 2: sigmoid gating; dot0 lives on lanes 0/16, dot1 on lanes 1/17 ----
    const float bias0 = gb[0];
    const float bias1 = gb[1];

    float ca8[8], cb8[8];   // valid on lanes 0 (M=0..7) and 16 (M=8..15)
#pragma unroll
    for (int r = 0; r < 8; ++r) {
        float d0 = acc[r];
        float d1 = __shfl_xor(d0, 1, 32);
        float z0 = d0 + bias0;
        float z1 = d1 + bias1;
        float s0 = 1.0f / (1.0f + expf(-z0));
        float s1 = 1.0f / (1.0f + expf(-z1));
        float g  = (z0 > 0.0f ? s0 : 0.0f) + (z1 > 0.0f ? s1 : 0.0f);
        float c  = (float)((int)(z0 > 0.0f) + (int)(z1 > 0.0f));
        ca8[r] = g;          // pre_x scale
        cb8[r] = c * g;      // post_x scale
    }

    // Broadcast the 16 per-token scalars to every lane.
    float ca16[16], cb16[16];
#pragma unroll
    for (int t = 0; t < TILE_M; ++t) {
        const int srcLane = (t >> 3) << 4;                 // 0 or 16
        ca16[t] = __shfl(ca8[t & 7], srcLane, 32);
        cb16[t] = __shfl(cb8[t & 7], srcLane, 32);
    }

    // ---- Phase 3: stream 16 rows: b128 load + 2x NT b128 store ----
    const size_t base = (size_t)token0 * D_MODEL;
    const float* __restrict__ xb  = x + base;
    float* __restrict__ prb = outPre + base;
    float* __restrict__ pob = outPost + base;

    for (int jj = 0; jj < D4 / 32; ++jj) {                 // 16 chunks per row
        const int col = (lane + jj * 32) * 4;              // float offset of this float4
#pragma unroll
        for (int t = 0; t < TILE_M; ++t) {                 // 16 rows, unrolled for MLP
            const int off = t * D_MODEL + col;
            v4f xv = *(const v4f*)(xb + off);
            __builtin_nontemporal_store(ca16[t] * xv, (v4f*)(prb + off));
            __builtin_nontemporal_store(cb16[t] * xv, (v4f*)(pob + off));
        }
    }
}

extern "C" void kernel_launch(void* const* d_in, const int* in_sizes, int n_in,
                              void* d_out, int out_size, void* d_ws, size_t ws_size,
                              hipStream_t stream) {
    const float* x  = (const float*)d_in[0];
    const float* gw = (const float*)d_in[1];
    const float* gb = (const float*)d_in[2];
    float* outPre  = (float*)d_out;
    float* outPost = outPre + (size_t)N_TOKENS * D_MODEL;

    dim3 grid(N_TOKENS / TILE_M);   // 1024 wave-tiles
    dim3 block(32);                 // one wave32 per workgroup
    hipLaunchKernelGGL(WeightedBranchRoute_kernel, grid, block, 0, stream,
                       x, gw, gb, outPre, outPost);
}